// VmfProductPosterior_180388626732
// MI455X (gfx1250) — compile-verified
//
#include <hip/hip_runtime.h>
#include <stdint.h>

typedef __attribute__((ext_vector_type(2))) float v2f;
typedef __attribute__((ext_vector_type(8))) float v8f;

#define N_VMFS 64
#define MDIM   5
#define PAIRS_PER_BLOCK 4
#define THREADS (PAIRS_PER_BLOCK * N_VMFS)   // 256 threads = 8 wave32

// ---------------- deterministic stateless RNG (PCG32) ----------------
__device__ __forceinline__ unsigned pcg32(unsigned long long& st) {
    unsigned long long old = st;
    st = old * 6364136223846793005ULL + 1442695040888963407ULL;
    unsigned xs  = (unsigned)(((old >> 18u) ^ old) >> 27u);
    unsigned rot = (unsigned)(old >> 59u);
    return (xs >> rot) | (xs << ((32u - rot) & 31u));
}
__device__ __forceinline__ float u01(unsigned long long& st) {
    // uniform in (0,1), never exactly 0 or 1
    return ((float)(pcg32(st) >> 8) + 0.5f) * (1.0f / 16777216.0f);
}

__global__ __launch_bounds__(THREADS)
void vmf_sample_kernel(const float* __restrict__ km,     // [B][64][5]
                       float* __restrict__ samples,      // [B][S][320]
                       float* __restrict__ logp,         // [B][S]
                       int nPairs, int S)
{
    __shared__ float s_km[N_VMFS * MDIM];   // 320 floats, staged kappa_mu slice
    __shared__ float s_logp[THREADS];       // 256 per-n log-prob terms

    const int tid   = threadIdx.x;
    const int gBase = blockIdx.x * PAIRS_PER_BLOCK;
    const int pair  = tid >> 6;             // 0..3
    const int n     = tid & 63;             // vmf index
    const int g     = gBase + pair;         // flat (b*S + s)
    const int bBlk  = gBase / S;            // batch row staged into LDS

    // ---- CDNA5 async global -> LDS staging of kappa_mu[bBlk] (320 dwords) ----
    {
        const float* gsrc = km + (size_t)bBlk * (N_VMFS * MDIM);
        unsigned lds_base = (unsigned)(uintptr_t)(&s_km[0]);   // low 32 bits = LDS offset
        unsigned long long ga0 = (unsigned long long)(uintptr_t)(gsrc + tid);
        unsigned off0 = lds_base + (unsigned)tid * 4u;
        asm volatile("global_load_async_to_lds_b32 %0, %1, off"
                     :: "v"(off0), "v"(ga0) : "memory");
        if (tid < (N_VMFS * MDIM - THREADS)) {  // 64 remaining dwords (waves 0-1, full EXEC)
            unsigned long long ga1 = (unsigned long long)(uintptr_t)(gsrc + THREADS + tid);
            unsigned off1 = lds_base + (unsigned)(THREADS + tid) * 4u;
            asm volatile("global_load_async_to_lds_b32 %0, %1, off"
                         :: "v"(off1), "v"(ga1) : "memory");
        }
        asm volatile("s_wait_asynccnt 0" ::: "memory");
    }
    __syncthreads();

    const bool active = (g < nPairs);
    const int  b = g / S;

    float k0 = 1.f, k1 = 0.f, k2 = 0.f, k3 = 0.f, k4 = 0.f;
    if (active) {
        if (b == bBlk) {
            k0 = s_km[n * 5 + 0]; k1 = s_km[n * 5 + 1]; k2 = s_km[n * 5 + 2];
            k3 = s_km[n * 5 + 3]; k4 = s_km[n * 5 + 4];
        } else {  // only possible if S % 4 != 0
            const float* p = km + ((size_t)b * N_VMFS + n) * MDIM;
            k0 = p[0]; k1 = p[1]; k2 = p[2]; k3 = p[3]; k4 = p[4];
        }
    }

    // ---- Wood (1994) envelope parameters (m = 5, mm1 = 4) ----
    float kap = sqrtf(k0*k0 + k1*k1 + k2*k2 + k3*k3 + k4*k4);
    float inv = 1.f / (kap + 1e-5f);
    float l0 = k0*inv, l1 = k1*inv, l2 = k2*inv, l3 = k3*inv, l4 = k4*inv;
    float cc     = sqrtf(4.f * kap * kap + 16.f);
    float b_true = (cc - 2.f * kap) * 0.25f;
    float b_app  = 1.f / kap;                         // mm1/(4*kap)
    float sm     = fminf(fmaxf(kap - 10.f, 0.f), 1.f);
    float bbv    = b_app * sm + b_true * (1.f - sm);
    float av     = (4.f + 2.f * kap + cc) * 0.25f;
    float dv     = 4.f * av * bbv / (1.f + bbv) - 5.5451774445f;  // - 4*log(4)

    // ---- early-exit rejection sampling; Beta(2,2) = median of 3 uniforms ----
    unsigned long long st =
        ((unsigned long long)(g * 64 + n) + 1ULL) * 0x9E3779B97F4A7C15ULL
        ^ 0x8A5CD789635D2DFFULL;
    (void)pcg32(st);  // warm-up mix

    float e_acc = -1.f, e_first = 0.5f;
    for (int kk = 0; kk < 64; ++kk) {
        float u1 = u01(st), u2 = u01(st), u3 = u01(st);
        float mx = fmaxf(u1, fmaxf(u2, u3));
        float mn = fminf(u1, fminf(u2, u3));
        float e  = u1 + u2 + u3 - mx - mn;            // ~ Beta(2,2)
        if (kk == 0) e_first = e;
        float uu = u01(st);
        float t  = 2.f * av * bbv / (1.f - (1.f - bbv) * e);
        if (4.f * __logf(t) - t + dv > __logf(uu)) { e_acc = e; break; }
    }
    if (e_acc < 0.f) e_acc = e_first;
    float w = (1.f - (1.f + bbv) * e_acc) / (1.f - (1.f - bbv) * e_acc);

    // ---- tangential direction: normalized 4-D Gaussian (Box-Muller) ----
    float ua = u01(st), ub = u01(st), uc = u01(st), ud = u01(st);
    float r1 = sqrtf(-2.f * __logf(ua)), r2 = sqrtf(-2.f * __logf(uc));
    float th1 = 6.28318530718f * ub,     th2 = 6.28318530718f * ud;
    float v0 = r1 * __cosf(th1), v1 = r1 * __sinf(th1);
    float v2 = r2 * __cosf(th2), v3 = r2 * __sinf(th2);
    float vn = rsqrtf(v0*v0 + v1*v1 + v2*v2 + v3*v3 + 1e-20f);
    v0 *= vn; v1 *= vn; v2 *= vn; v3 *= vn;
    float w_ = sqrtf(fmaxf(1.f - w * w, 1e-10f));
    float x0 = w, x1 = w_*v0, x2 = w_*v1, x3 = w_*v2, x4 = w_*v3;

    // ---- Householder rotation e1 -> loc ----
    float h0 = 1.f - l0, h1 = -l1, h2 = -l2, h3 = -l3, h4 = -l4;
    float hi = 1.f / (sqrtf(h0*h0 + h1*h1 + h2*h2 + h3*h3 + h4*h4) + 1e-5f);
    h0 *= hi; h1 *= hi; h2 *= hi; h3 *= hi; h4 *= hi;
    float dxu = x0*h0 + x1*h1 + x2*h2 + x3*h3 + x4*h4;
    float z0 = x0 - 2.f*dxu*h0, z1 = x1 - 2.f*dxu*h1, z2 = x2 - 2.f*dxu*h2;
    float z3 = x3 - 2.f*dxu*h3, z4 = x4 - 2.f*dxu*h4;

    // ---- per-n log prob term ----
    float dlz  = l0*z0 + l1*z1 + l2*z2 + l3*z3 + l4*z4;
    float em2x = __expf(-2.f * kap);
    float live = -0.5f * __logf(6.28318530718f * kap)
               + __logf((1.f + em2x) - (1.f - em2x) / kap);   // log I_1.5(k) e^{-k}
    float log_norm = -(1.5f * __logf(kap) - 4.5946926682f - (kap + live)); // 2.5*log(2pi)
    float lp = kap * dlz - log_norm;

    if (active) {
        float* o = samples + (size_t)g * (N_VMFS * MDIM) + n * MDIM;
        o[0] = z0; o[1] = z1; o[2] = z2; o[3] = z3; o[4] = z4;
    }
    s_logp[tid] = active ? lp : 0.f;
    __syncthreads();

    // ---- WMMA f32 16x16x4 reduction: 4 chained D = A*1 + C over 256 LDS values ----
    if (tid < 32) {                       // wave 0, EXEC all ones
        v8f acc = {0.f, 0.f, 0.f, 0.f, 0.f, 0.f, 0.f, 0.f};
        v2f bone; bone.x = 1.f; bone.y = 1.f;   // B = ones (layout-invariant)
        int r   = tid & 15;               // matrix row = pair*4 + chunk
        int p   = r >> 2;
        int cch = r & 3;
        int kof = (tid < 16) ? 0 : 2;     // A 16x4: lanes 0-15 hold K0/K1, 16-31 K2/K3
        const float* base = &s_logp[p * 64 + cch * 16 + kof];
#pragma unroll
        for (int t4 = 0; t4 < 4; ++t4) {
            v2f aV; aV.x = base[t4 * 4 + 0]; aV.y = base[t4 * 4 + 1];
            acc = __builtin_amdgcn_wmma_f32_16x16x4_f32(
                      false, aV, false, bone, (short)0, acc, false, false);
        }
        // C/D layout: lanes 0-15 -> rows 0-7 in acc[0..7]; lanes 16-31 -> rows 8-15
        float sA = acc[0] + acc[1] + acc[2] + acc[3];   // pair 0 (lane 0) / pair 2 (lane 16)
        float sB = acc[4] + acc[5] + acc[6] + acc[7];   // pair 1 (lane 0) / pair 3 (lane 16)
        if (tid == 0) {
            if (gBase + 0 < nPairs) logp[gBase + 0] = sA;
            if (gBase + 1 < nPairs) logp[gBase + 1] = sB;
        } else if (tid == 16) {
            if (gBase + 2 < nPairs) logp[gBase + 2] = sA;
            if (gBase + 3 < nPairs) logp[gBase + 3] = sB;
        }
    }
}

extern "C" void kernel_launch(void* const* d_in, const int* in_sizes, int n_in,
                              void* d_out, int out_size, void* d_ws, size_t ws_size,
                              hipStream_t stream) {
    (void)n_in; (void)d_ws; (void)ws_size;
    const float* km = (const float*)d_in[0];            // [B][64][5]
    int Bsz = in_sizes[0] / (N_VMFS * MDIM);            // 1024
    int S   = out_size / (Bsz * (N_VMFS * MDIM + 1));   // 16 (samples + logp packed)
    int nPairs = Bsz * S;
    if (nPairs <= 0) return;
    float* out_samples = (float*)d_out;                 // nPairs * 320
    float* out_logp    = out_samples + (size_t)nPairs * (N_VMFS * MDIM);
    int blocks = (nPairs + PAIRS_PER_BLOCK - 1) / PAIRS_PER_BLOCK;
    vmf_sample_kernel<<<blocks, THREADS, 0, stream>>>(km, out_samples, out_logp,
                                                      nPairs, S);
}